// GCNN_42863773614285
// MI455X (gfx1250) — compile-verified
//
#include <hip/hip_runtime.h>
#include <math.h>

typedef float v2f __attribute__((ext_vector_type(2)));
typedef float v8f __attribute__((ext_vector_type(8)));

#define LDS_STRIDE 132   // 128 + 4 pad: avoids 64-bank conflicts on column reads

// ---------------------------------------------------------------- degree prep
__global__ void k_init_deg(float* __restrict__ deg, int N) {
  int i = blockIdx.x * blockDim.x + threadIdx.x;
  if (i < N) deg[i] = 1.0f;  // self-loop contributes 1
}

__global__ void k_deg_accum(const int* __restrict__ dst, float* __restrict__ deg, int E) {
  int e = blockIdx.x * blockDim.x + threadIdx.x;
  if (e < E)
    __hip_atomic_fetch_add(&deg[dst[e]], 1.0f, __ATOMIC_RELAXED, __HIP_MEMORY_SCOPE_AGENT);
}

__global__ void k_rsqrt(float* __restrict__ deg, int N) {
  int i = blockIdx.x * blockDim.x + threadIdx.x;
  if (i < N) deg[i] = rsqrtf(deg[i]);  // deg >= 1 always (self-loop)
}

// ---------------------------------------------------------------- GEMM1: h1 = x @ W1  (fp32 WMMA)
// Block = 128 threads (4 waves). Each block: 16 rows of x; each wave: 16-col slab of W1.
__global__ void k_gemm1_wmma(const float* __restrict__ x, const float* __restrict__ W1,
                             float* __restrict__ h1, int N) {
  __shared__ float sx[16 * LDS_STRIDE];
  const int tid  = threadIdx.x;
  const int row0 = blockIdx.x * 16;

  // cooperative load of the contiguous 16x128 tile into LDS (padded stride)
  const size_t gbase = (size_t)row0 * 128;
  const size_t gmax  = (size_t)N * 128;
  for (int g = tid * 4; g < 16 * 128; g += 128 * 4) {
    if (gbase + (size_t)g + 3 < gmax) {
      float4 v = *(const float4*)(x + gbase + g);
      int m = g >> 7, kc = g & 127;
      *(float4*)(&sx[m * LDS_STRIDE + kc]) = v;
    }
  }
  __syncthreads();

  const int lane  = tid & 31;
  const int wave  = tid >> 5;
  const int mrow  = lane & 15;        // A: M = lane%16
  const int khalf = lane >> 4;        // A/B: lanes 16-31 carry K+2 pair
  const int ncol  = wave * 16 + (lane & 15);  // B/C: N = lane%16 within slab

  v8f c = {0.f, 0.f, 0.f, 0.f, 0.f, 0.f, 0.f, 0.f};

#if __has_builtin(__builtin_amdgcn_wmma_f32_16x16x4_f32)
  for (int k = 0; k < 128; k += 4) {
    const int ka = k + 2 * khalf;
    v2f a, b;
    a[0] = sx[mrow * LDS_STRIDE + ka];
    a[1] = sx[mrow * LDS_STRIDE + ka + 1];
    b[0] = W1[ka * 64 + ncol];
    b[1] = W1[(ka + 1) * 64 + ncol];
    // D = A(16x4) * B(4x16) + C  — fp32, exact match to reference math
    c = __builtin_amdgcn_wmma_f32_16x16x4_f32(false, a, false, b, (short)0, c,
                                              false, false);
  }
#else
#warning "wmma f32 16x16x4 builtin unavailable - scalar fallback emitted"
  for (int v = 0; v < 8; ++v) {
    float acc = 0.f;
    const int m = v + 8 * khalf;
    for (int k = 0; k < 128; ++k) acc += sx[m * LDS_STRIDE + k] * W1[k * 64 + ncol];
    c[v] = acc;
  }
#endif

  // C layout: VGPR v -> row (v + 8*khalf), col = ncol
  for (int v = 0; v < 8; ++v) {
    int row = row0 + v + 8 * khalf;
    if (row < N) h1[(size_t)row * 64 + ncol] = c[v];
  }
}

// ---------------------------------------------------------------- agg1 = broadcast b1
__global__ void k_init_agg1(float* __restrict__ agg1, const float* __restrict__ b1, int total) {
  int i = blockIdx.x * blockDim.x + threadIdx.x;
  if (i < total) agg1[i] = b1[i & 63];
}

// ---------------------------------------------------------------- scatter1: agg1[dst] += norm * h1[src]
// 16 threads per edge, each owns a float4 chunk of the 64-wide row.
__global__ void k_scatter1(const int* __restrict__ src, const int* __restrict__ dst,
                           const float* __restrict__ dinv, const float* __restrict__ h1,
                           float* __restrict__ agg1, int E, int N) {
  int gid = blockIdx.x * blockDim.x + threadIdx.x;
  int e = gid >> 4;
  if (e >= E + N) return;
  int cc = (gid & 15) * 4;
  int s, d;
  if (e < E) { s = src[e]; d = dst[e]; } else { s = e - E; d = s; }
  float nrm = dinv[s] * dinv[d];
  float4 v = *(const float4*)(h1 + (size_t)s * 64 + cc);
  float* base = agg1 + (size_t)d * 64 + cc;
  __hip_atomic_fetch_add(base + 0, nrm * v.x, __ATOMIC_RELAXED, __HIP_MEMORY_SCOPE_AGENT);
  __hip_atomic_fetch_add(base + 1, nrm * v.y, __ATOMIC_RELAXED, __HIP_MEMORY_SCOPE_AGENT);
  __hip_atomic_fetch_add(base + 2, nrm * v.z, __ATOMIC_RELAXED, __HIP_MEMORY_SCOPE_AGENT);
  __hip_atomic_fetch_add(base + 3, nrm * v.w, __ATOMIC_RELAXED, __HIP_MEMORY_SCOPE_AGENT);
}

// ---------------------------------------------------------------- GEMM2: h2 = leakyrelu(agg1) @ W2   (64x4, tiny)
__global__ void k_gemm2(const float* __restrict__ agg1, const float* __restrict__ W2,
                        float* __restrict__ h2, int N) {
  int i = blockIdx.x * blockDim.x + threadIdx.x;
  if (i >= N) return;
  const float* row = agg1 + (size_t)i * 64;
  float a0 = 0.f, a1 = 0.f, a2 = 0.f, a3 = 0.f;
  for (int k = 0; k < 64; ++k) {
    float v = row[k];
    v = v > 0.f ? v : 0.01f * v;  // leaky_relu(0.01), dropout is identity in eval
    float4 w = *(const float4*)(W2 + k * 4);
    a0 += v * w.x; a1 += v * w.y; a2 += v * w.z; a3 += v * w.w;
  }
  float4 o = {a0, a1, a2, a3};
  *(float4*)(h2 + (size_t)i * 4) = o;
}

// ---------------------------------------------------------------- out = broadcast b2
__global__ void k_init_out(float* __restrict__ out, const float* __restrict__ b2, int total) {
  int i = blockIdx.x * blockDim.x + threadIdx.x;
  if (i < total) out[i] = b2[i & 3];
}

// ---------------------------------------------------------------- scatter2: out[dst] += norm * h2[src]
__global__ void k_scatter2(const int* __restrict__ src, const int* __restrict__ dst,
                           const float* __restrict__ dinv, const float* __restrict__ h2,
                           float* __restrict__ out, int E, int N) {
  int e = blockIdx.x * blockDim.x + threadIdx.x;
  if (e >= E + N) return;
  int s, d;
  if (e < E) { s = src[e]; d = dst[e]; } else { s = e - E; d = s; }
  float nrm = dinv[s] * dinv[d];
  float4 v = *(const float4*)(h2 + (size_t)s * 4);
  float* base = out + (size_t)d * 4;
  __hip_atomic_fetch_add(base + 0, nrm * v.x, __ATOMIC_RELAXED, __HIP_MEMORY_SCOPE_AGENT);
  __hip_atomic_fetch_add(base + 1, nrm * v.y, __ATOMIC_RELAXED, __HIP_MEMORY_SCOPE_AGENT);
  __hip_atomic_fetch_add(base + 2, nrm * v.z, __ATOMIC_RELAXED, __HIP_MEMORY_SCOPE_AGENT);
  __hip_atomic_fetch_add(base + 3, nrm * v.w, __ATOMIC_RELAXED, __HIP_MEMORY_SCOPE_AGENT);
}

// ---------------------------------------------------------------- per-node softmax over 4 logits (in place)
__global__ void k_softmax4(float* __restrict__ out, int N) {
  int i = blockIdx.x * blockDim.x + threadIdx.x;
  if (i >= N) return;
  float4 v = *(float4*)(out + (size_t)i * 4);
  float m = fmaxf(fmaxf(v.x, v.y), fmaxf(v.z, v.w));
  float e0 = __expf(v.x - m), e1 = __expf(v.y - m), e2 = __expf(v.z - m), e3 = __expf(v.w - m);
  float inv = 1.0f / (e0 + e1 + e2 + e3);
  float4 o = {e0 * inv, e1 * inv, e2 * inv, e3 * inv};
  *(float4*)(out + (size_t)i * 4) = o;
}

// ----------------------------------------------------------------------------
extern "C" void kernel_launch(void* const* d_in, const int* in_sizes, int n_in,
                              void* d_out, int out_size, void* d_ws, size_t ws_size,
                              hipStream_t stream) {
  const float* x  = (const float*)d_in[0];   // [N,128]
  const float* W1 = (const float*)d_in[1];   // [128,64]
  const float* b1 = (const float*)d_in[2];   // [64]
  const float* W2 = (const float*)d_in[3];   // [64,4]
  const float* b2 = (const float*)d_in[4];   // [4]
  const int*   ei = (const int*)d_in[5];     // [2,E] (int32: jax x64 disabled)
  float* out = (float*)d_out;                // [N,4]

  const int N = in_sizes[0] / 128;
  const int E = in_sizes[5] / 2;
  const int* src = ei;
  const int* dst = ei + E;

  // workspace layout (floats): deg/dinv[N] | h1[N*64] | agg1[N*64] | h2[N*4]  (~53 MB)
  float* wsf  = (float*)d_ws;
  float* deg  = wsf;
  float* h1   = deg + N;
  float* agg1 = h1 + (size_t)N * 64;
  float* h2   = agg1 + (size_t)N * 64;

  const int B = 256;
  // 1-3: symmetric normalization dinv = rsqrt(deg)
  k_init_deg<<<(N + B - 1) / B, B, 0, stream>>>(deg, N);
  k_deg_accum<<<(E + B - 1) / B, B, 0, stream>>>(dst, deg, E);
  k_rsqrt<<<(N + B - 1) / B, B, 0, stream>>>(deg, N);
  // 4: h1 = x @ W1 via v_wmma_f32_16x16x4_f32
  k_gemm1_wmma<<<(N + 15) / 16, 128, 0, stream>>>(x, W1, h1, N);
  // 5-6: agg1 = b1 + scatter(norm * h1[src])
  k_init_agg1<<<((N * 64) + B - 1) / B, B, 0, stream>>>(agg1, b1, N * 64);
  {
    long long total = (long long)(E + N) * 16;
    k_scatter1<<<(int)((total + B - 1) / B), B, 0, stream>>>(src, dst, deg, h1, agg1, E, N);
  }
  // 7: h2 = leakyrelu(agg1) @ W2
  k_gemm2<<<(N + B - 1) / B, B, 0, stream>>>(agg1, W2, h2, N);
  // 8-9: out = b2 + scatter(norm * h2[src])
  k_init_out<<<((N * 4) + B - 1) / B, B, 0, stream>>>(out, b2, N * 4);
  k_scatter2<<<(E + N + B - 1) / B, B, 0, stream>>>(src, dst, deg, h2, out, E, N);
  // 10: row softmax
  k_softmax4<<<(N + B - 1) / B, B, 0, stream>>>(out, N);
}